// UFLNeoHookeanModel_47141561041120
// MI455X (gfx1250) — compile-verified
//
#include <hip/hip_runtime.h>

// CDNA5 / gfx1250 Neo-Hookean strain-energy reduction.
// Core contraction grad_u[b,q,i,j] = sum_n u[b,elem[e,n],i] * dN[e,q,n,j]
// is mapped onto V_WMMA_F32_16X16X4_F32 (K=4 == NPE, full f32 precision):
//   A[m=b*4+i][k=n]   (rows with i==3 zero-padded)
//   B[k=n][n'=q*4+j]  (cols with j==3 zero-padded)
//   D[m][n'] = grad_u[b,q,i,j]   -> one WMMA computes 4 batches x 4 quads x 3x3.

typedef __attribute__((ext_vector_type(2))) float v2f;
typedef __attribute__((ext_vector_type(8))) float v8f;

namespace {
constexpr float kMu    = 10000.0f / (2.0f * (1.0f + 0.3f));            // 3846.1538
constexpr float kLmbda = 10000.0f * 0.3f / ((1.0f + 0.3f) * 0.4f);     // 5769.2307
constexpr float kEps   = 1e-10f;
constexpr int   kWaves = 8;
constexpr int   kBlock = kWaves * 32;
}

__global__ void nh_zero(float* out, int n) {
  int i = blockIdx.x * blockDim.x + threadIdx.x;
  if (i < n) out[i] = 0.0f;
}

__global__ __launch_bounds__(kBlock)
void nh_energy_wmma(const float* __restrict__ u,
                    const int*   __restrict__ elem,
                    const float* __restrict__ dN,
                    const float* __restrict__ detJ,
                    const float* __restrict__ qw,
                    float* __restrict__ out,
                    int n_elem, int n_nodes) {
  // Per-wave scratch: two 16x16 f32 D tiles + 32-float reduction slab.
  __shared__ float sD[kWaves][2][16][16];
  __shared__ float sR[kWaves][32];

  const int lane = threadIdx.x & 31;
  const int wv   = threadIdx.x >> 5;
  const int t    = lane & 15;     // A row / B col / D col index
  const int hi   = lane >> 4;     // lane half: selects K pair (and element in epilogue)
  const int k0   = hi * 2;        // this lane holds K = k0, k0+1 for both A and B
  const int ab   = t >> 2;        // A row -> batch b
  const int ai   = t & 3;         // A row -> i (3 => zero pad row)
  const int bq   = t >> 2;        // B col -> quad q
  const int bj   = t & 3;         // B col -> j (3 => zero pad col)
  const int eb   = t >> 2;        // epilogue batch
  const int eq   = t & 3;         // epilogue quad

  const float qwv = qw[eq];                        // quad weight for this lane's q
  const size_t ustride = (size_t)n_nodes * 3;

  const long long gw = ((long long)blockIdx.x * kBlock + threadIdx.x) >> 5;
  const long long nw = (long long)gridDim.x * kWaves;
  const long long npairs = (n_elem + 1) >> 1;

  float acc = 0.0f;

  for (long long p = gw; p < npairs; p += nw) {
    const int e0 = (int)(p << 1);

    // Prefetch next stride's dN block (gfx1250 global_prefetch_b8).
    if (p + nw < npairs) {
      __builtin_prefetch(dN + ((size_t)(p + nw) << 1) * 48, 0, 1);
    }

#pragma unroll
    for (int s = 0; s < 2; ++s) {
      const int ee  = e0 + s;
      const int eec = (ee < n_elem) ? ee : (n_elem - 1);   // uniform clamp (keeps EXEC full)

      const int idx0 = elem[eec * 4 + k0];
      const int idx1 = elem[eec * 4 + k0 + 1];

      v2f a, b;
      a.x = (ai < 3) ? u[(size_t)ab * ustride + (size_t)idx0 * 3 + ai] : 0.0f;
      a.y = (ai < 3) ? u[(size_t)ab * ustride + (size_t)idx1 * 3 + ai] : 0.0f;

      const float* dNe = dN + (size_t)eec * 48;            // (q,n,j) block, 48 f32
      b.x = (bj < 3) ? dNe[bq * 12 + k0 * 3 + bj]       : 0.0f;
      b.y = (bj < 3) ? dNe[bq * 12 + (k0 + 1) * 3 + bj] : 0.0f;

      v8f c = {0.0f, 0.0f, 0.0f, 0.0f, 0.0f, 0.0f, 0.0f, 0.0f};
      // D = A x B + 0 : all grad_u for this element, exact f32.
      v8f d = __builtin_amdgcn_wmma_f32_16x16x4_f32(
          false, a, false, b, (short)0, c, false, false);

      // Spill D tile: VGPR r holds row m = r + 8*hi, col n = t.
#pragma unroll
      for (int r = 0; r < 8; ++r)
        sD[wv][s][r + 8 * hi][t] = d[r];
    }
    // Same-wave LDS ops are in-order; fence only the scheduler.
    __builtin_amdgcn_wave_barrier();

    // Epilogue: lane -> (element = e0+hi, batch eb, quad eq).
    const int ee   = e0 + hi;
    const float on = (ee < n_elem) ? 1.0f : 0.0f;
    const int eec  = (ee < n_elem) ? ee : (n_elem - 1);

    float F[3][3];
#pragma unroll
    for (int i2 = 0; i2 < 3; ++i2)
#pragma unroll
      for (int j2 = 0; j2 < 3; ++j2)
        F[i2][j2] = sD[wv][hi][eb * 4 + i2][eq * 4 + j2] + ((i2 == j2) ? 1.0f : 0.0f);

    const float Jd = F[0][0] * (F[1][1] * F[2][2] - F[1][2] * F[2][1])
                   - F[0][1] * (F[1][0] * F[2][2] - F[1][2] * F[2][0])
                   + F[0][2] * (F[1][0] * F[2][1] - F[1][1] * F[2][0]);

    float IC = 0.0f;
#pragma unroll
    for (int i2 = 0; i2 < 3; ++i2)
#pragma unroll
      for (int j2 = 0; j2 < 3; ++j2)
        IC = fmaf(F[i2][j2], F[i2][j2], IC);

    const float lj = __logf(fmaxf(Jd, kEps));
    const float W  = 0.5f * kMu * (IC - 3.0f - 2.0f * lj)
                   + 0.25f * kLmbda * (Jd * Jd - 1.0f - 2.0f * lj);

    acc += on * W * detJ[(size_t)eec * 4 + eq] * qwv;
    __builtin_amdgcn_wave_barrier();
  }

  // Wave-local reduction: lanes {4b..4b+3} and {16+4b..16+4b+3} carry batch b.
  sR[wv][lane] = acc;
  __builtin_amdgcn_wave_barrier();
  if (lane < 4) {
    float s = 0.0f;
#pragma unroll
    for (int h = 0; h < 2; ++h)
#pragma unroll
      for (int k = 0; k < 4; ++k)
        s += sR[wv][h * 16 + lane * 4 + k];
    atomicAdd(&out[lane], s);
  }
}

extern "C" void kernel_launch(void* const* d_in, const int* in_sizes, int n_in,
                              void* d_out, int out_size, void* d_ws, size_t ws_size,
                              hipStream_t stream) {
  const float* u    = (const float*)d_in[0];   // (B, N_NODES, 3)
  const int*   el   = (const int*)  d_in[1];   // (N_ELEM, 4)
  const float* dN   = (const float*)d_in[2];   // (N_ELEM, 4, 4, 3)
  const float* detJ = (const float*)d_in[3];   // (N_ELEM, 4)
  const float* qw   = (const float*)d_in[4];   // (4,)
  float* out = (float*)d_out;                  // (B,) f32

  const int n_elem  = in_sizes[1] / 4;
  const int Bsz     = out_size > 0 ? out_size : 4;
  const int n_nodes = in_sizes[0] / (3 * Bsz);

  nh_zero<<<1, 64, 0, stream>>>(out, out_size);

  const long long npairs = (n_elem + 1) / 2;
  long long blocks = (npairs + kWaves - 1) / kWaves;
  if (blocks > 2048) blocks = 2048;
  if (blocks < 1) blocks = 1;
  nh_energy_wmma<<<(int)blocks, kBlock, 0, stream>>>(u, el, dN, detJ, qw, out,
                                                     n_elem, n_nodes);
}